// CompactBilinearPooling_24455543783924
// MI455X (gfx1250) — compile-verified
//
#include <hip/hip_runtime.h>

typedef __attribute__((ext_vector_type(2))) float v2f;
typedef __attribute__((ext_vector_type(8))) float v8f;

#define B_DIM   16
#define C_DIM   512
#define HW_DIM  196
#define D_DIM   8000

// ---------------------------------------------------------------------------
// Kernel 1: recover count-sketch (h, s) arrays from the dense sketch matrices.
// Each row of S has exactly one nonzero (±1), so exactly one thread per row
// writes -> no atomics needed. This is the HBM-dominant phase (~33 MB scan).
// ---------------------------------------------------------------------------
__global__ __launch_bounds__(256)
void cbp_extract_sketch(const float* __restrict__ S1, const float* __restrict__ S2,
                        int*  __restrict__ h1, float* __restrict__ s1,
                        int*  __restrict__ h2, float* __restrict__ s2) {
  long idx = (long)blockIdx.x * blockDim.x + threadIdx.x;
  const long per = (long)C_DIM * D_DIM;
  if (idx >= 2 * per) return;
  const int  m   = idx >= per;
  const long rem = idx - (long)m * per;
  const int  c   = (int)(rem / D_DIM);
  const int  d   = (int)(rem - (long)c * D_DIM);
  const float v  = m ? S2[rem] : S1[rem];
  if (v != 0.0f) {
    if (m) { h2[c] = d; s2[c] = v; }
    else   { h1[c] = d; s1[c] = v; }
  }
}

// ---------------------------------------------------------------------------
// Kernel 2: fused Gram (WMMA f32 16x16x4) + count-sketch scatter.
// grid = (16 i-blocks of 32 rows, 16 batches), 256 threads = 8 wave32s.
// Each wave computes 8 tiles of G[b] = X_b * X_b^T (X_b: [512 x 196] row-major,
// which is exactly the NCHW layout of x[b]), then scatters each 16x16 tile
// into an LDS-private 8000-bin histogram via ds_add_f32.
// ---------------------------------------------------------------------------
__global__ __launch_bounds__(256)
void cbp_gram_sketch(const float* __restrict__ x,
                     const int*  __restrict__ h1, const float* __restrict__ s1,
                     const int*  __restrict__ h2, const float* __restrict__ s2,
                     float* __restrict__ out) {
  __shared__ float acc[D_DIM];            // 32 KB of the 320 KB WGP LDS

  const int b    = blockIdx.y;
  const int iblk = blockIdx.x;            // 32 rows of G per block
  const int tid  = threadIdx.x;
  const int wave = tid >> 5;              // wave32
  const int lane = tid & 31;

  for (int d = tid; d < D_DIM; d += 256) acc[d] = 0.0f;
  __syncthreads();

  const float* X     = x + (long)b * C_DIM * HW_DIM;
  const int    lrow  = lane & 15;             // M (or N) row within tile
  const int    klane = (lane >> 4) << 1;      // lanes 16-31 hold K+2,K+3

  // 2 i-tiles x 32 j-tiles = 64 tiles per block, 8 per wave
  for (int t = 0; t < 8; ++t) {
    const int tile = wave * 8 + t;
    const int it   = tile >> 5;               // 0..1
    const int jt   = tile & 31;               // 0..31
    const int i0   = iblk * 32 + it * 16;
    const int j0   = jt * 16;

    const float* Arow = X + (long)(i0 + lrow) * HW_DIM;
    const float* Brow = X + (long)(j0 + lrow) * HW_DIM;
    __builtin_prefetch(Brow, 0, 0);           // global_prefetch_b8

    v8f c = {};
#pragma unroll 7
    for (int k0 = 0; k0 < HW_DIM; k0 += 4) {  // 49 chained WMMAs, K=196
      v2f a  = *(const v2f*)(Arow + k0 + klane);
      v2f bb = *(const v2f*)(Brow + k0 + klane);
      // D = A(16x4,f32) * B(4x16,f32) + C(16x16,f32)
      c = __builtin_amdgcn_wmma_f32_16x16x4_f32(
              /*neg_a=*/false, a, /*neg_b=*/false, bb,
              /*c_mod=*/(short)0, c, /*reuse_a=*/false, /*reuse_b=*/false);
    }

    // Scatter the tile: lane holds column j = j0 + (lane&15); VGPR r holds
    // row i0 + r (+8 for lanes 16-31), per the 16x16 f32 C/D layout.
    const int   j   = j0 + lrow;
    const int   hj  = h2[j];
    const float sj  = s2[j];
    const int   ib  = i0 + ((lane >> 4) << 3);
#pragma unroll
    for (int r = 0; r < 8; ++r) {
      const int   i = ib + r;
      int d = h1[i] + hj;
      d = (d >= D_DIM) ? d - D_DIM : d;
      atomicAdd(&acc[d], s1[i] * sj * c[r]);  // ds_add_f32 (no return)
    }
  }

  __syncthreads();
  float* ob = out + (long)b * D_DIM;
  for (int d = tid; d < D_DIM; d += 256)
    unsafeAtomicAdd(&ob[d], acc[d]);          // global_atomic_add_f32
}

// ---------------------------------------------------------------------------
extern "C" void kernel_launch(void* const* d_in, const int* in_sizes, int n_in,
                              void* d_out, int out_size, void* d_ws, size_t ws_size,
                              hipStream_t stream) {
  const float* x  = (const float*)d_in[0];   // [16,512,14,14] f32
  const float* S1 = (const float*)d_in[1];   // [512,8000] f32
  const float* S2 = (const float*)d_in[2];   // [512,8000] f32
  float* out = (float*)d_out;                // [16,8000] f32

  char*  ws = (char*)d_ws;                   // 8 KB of scratch used
  int*   h1 = (int*)(ws);
  int*   h2 = (int*)(ws + C_DIM * 4);
  float* s1 = (float*)(ws + C_DIM * 8);
  float* s2 = (float*)(ws + C_DIM * 12);

  hipMemsetAsync(d_out, 0, (size_t)out_size * sizeof(float), stream);

  const long total = 2L * C_DIM * D_DIM;
  const int  blks  = (int)((total + 255) / 256);
  cbp_extract_sketch<<<blks, 256, 0, stream>>>(S1, S2, h1, s1, h2, s2);

  dim3 grid(16, B_DIM);                      // 256 workgroups, 2048 wave32s
  cbp_gram_sketch<<<grid, 256, 0, stream>>>(x, h1, s1, h2, s2, out);
}